// SEWHTGate_13116830121985
// MI455X (gfx1250) — compile-verified
//
#include <hip/hip_runtime.h>
#include <hip/hip_bf16.h>

#define B_ 8
#define N_ 4096
#define D_ 768
#define P_ 1024

typedef float v2f __attribute__((ext_vector_type(2)));
typedef float v4f __attribute__((ext_vector_type(4)));
typedef float v8f __attribute__((ext_vector_type(8)));

// ---------------------------------------------------------------------------
// Kernel 0: zero the mean accumulator (8*768 floats)
// ---------------------------------------------------------------------------
__global__ void sewht_zero_kernel(float* __restrict__ c) {
    int i = blockIdx.x * 256 + threadIdx.x;
    if (i < B_ * D_) c[i] = 0.0f;
}

// ---------------------------------------------------------------------------
// Kernel 1: c[b][d] = mean over N of x[b][n][d]
// grid = (3 d-chunks, 16 n-chunks, 8 batches), 256 threads.
// Consecutive threads read consecutive d -> fully coalesced rows.
// Default (RT) loads intentionally populate L2: x (100 MB) fits in the
// 192 MB L2 and is re-read by the scaling pass.
// ---------------------------------------------------------------------------
__global__ __launch_bounds__(256) void sewht_mean_kernel(
        const float* __restrict__ x, float* __restrict__ c) {
    int d  = blockIdx.x * 256 + threadIdx.x;   // 0..767
    int b  = blockIdx.z;                       // 0..7
    int n0 = blockIdx.y * 256;                 // n-chunk base
    const float* p = x + ((size_t)b * N_ + n0) * D_ + d;
    float s = 0.0f;
    #pragma unroll 8
    for (int n = 0; n < 256; ++n) s += p[(size_t)n * D_];
    atomicAdd(&c[b * D_ + d], s * (1.0f / (float)N_));
}

// ---------------------------------------------------------------------------
// Kernel 2: pad c to 1024, fast WHT over 1024 (10 LDS butterfly stages),
// scale by 1/1024, multiply by wht_weight, store zero-padded 16x1024
// A-matrix (rows 8..15 = 0) for the WMMA gate GEMM.
// Single block, 1024 threads, 32 KB LDS.
// ---------------------------------------------------------------------------
__global__ __launch_bounds__(1024) void sewht_wht_kernel(
        const float* __restrict__ c, const float* __restrict__ w,
        float* __restrict__ cw) {
    __shared__ float lds[B_][P_];
    int t = threadIdx.x;   // 0..1023

    #pragma unroll
    for (int b = 0; b < B_; ++b)
        lds[b][t] = (t < D_) ? c[b * D_ + t] : 0.0f;
    __syncthreads();

    // 10 stages; 8*512 = 4096 butterflies/stage, 4 per thread.
    for (int st = 0; st < 10; ++st) {
        int s = 1 << st;
        #pragma unroll
        for (int it = 0; it < 4; ++it) {
            int f = it * 1024 + t;               // 0..4095
            int b = f >> 9;                      // /512
            int p = f & 511;
            int i = ((p >> st) << (st + 1)) | (p & (s - 1));
            int j = i + s;
            float a  = lds[b][i];
            float bb = lds[b][j];
            lds[b][i] = a + bb;
            lds[b][j] = a - bb;
        }
        __syncthreads();
    }

    float wk = w[t] * (1.0f / (float)P_);
    #pragma unroll
    for (int b = 0; b < B_; ++b)
        cw[b * P_ + t] = lds[b][t] * wk;
    #pragma unroll
    for (int b = B_; b < 16; ++b)
        cw[b * P_ + t] = 0.0f;                   // zero-pad M rows 8..15
}

// ---------------------------------------------------------------------------
// Kernel 3: gate[b][d] = sigmoid( sum_k cw[b][k]*gate_w[d][k] + gate_b[d] )
// via V_WMMA_F32_16X16X4_F32.  One wave per 16-wide N tile (48 blocks).
//
// 32-bit A 16x4 layout: lanes 0-15 hold M=lane with K={0,1} in v[0],v[1];
// lanes 16-31 hold M=lane-16 with K={2,3}.  B 4x16 mirrors this with N on
// the lane axis.  So each lane loads float2 at (row_base + k + koff),
// koff = 2*(lane>>4), for both operands.
// D layout: lanes 0-15, VGPR m -> D[M=m][N=lane] for m=0..7 (our 8 batches).
// ---------------------------------------------------------------------------
__global__ __launch_bounds__(32) void sewht_gate_kernel(
        const float* __restrict__ cw, const float* __restrict__ gw,
        const float* __restrict__ gb, float* __restrict__ gate) {
    int lane = threadIdx.x;                // 0..31 (wave32)
    int tile = blockIdx.x;                 // 0..47 -> N columns [16*tile, +16)
    int r    = lane & 15;
    int koff = (lane >> 4) << 1;           // 0 or 2

    const float* ap = cw + r * P_ + koff;                       // A: M=r
    const float* bp = gw + (size_t)(tile * 16 + r) * P_ + koff; // B: N=r

    v8f acc = {0.f, 0.f, 0.f, 0.f, 0.f, 0.f, 0.f, 0.f};
    #pragma unroll 4
    for (int k = 0; k < P_; k += 4) {
        v2f a = *(const v2f*)(ap + k);
        v2f b = *(const v2f*)(bp + k);
        acc = __builtin_amdgcn_wmma_f32_16x16x4_f32(
            /*neg_a=*/false, a, /*neg_b=*/false, b,
            /*c_mod=*/(short)0, acc, /*reuse_a=*/false, /*reuse_b=*/false);
    }

    if (lane < 16) {                       // lanes 0-15 hold M=0..7 in v0..v7
        int d = tile * 16 + lane;
        float bias = gb[d];
        #pragma unroll
        for (int m = 0; m < B_; ++m) {
            float v = acc[m] + bias;
            gate[m * D_ + d] = 1.0f / (1.0f + __expf(-v));
        }
    }
}

// ---------------------------------------------------------------------------
// Kernel 4: out[b][n][d] = x[b][n][d] * gate[b][d], float4-vectorized.
// 8*4096*192 = 6,291,456 float4s = 24576 blocks * 256 threads exactly.
// x loads use default RT policy (hits L2 lines resident from the mean pass);
// out stores are non-temporal so the 100 MB output stream does not evict
// x from the 192 MB L2 mid-pass.
// ---------------------------------------------------------------------------
__global__ __launch_bounds__(256) void sewht_scale_kernel(
        const v4f* __restrict__ x4, const float* __restrict__ gate,
        v4f* __restrict__ out4) {
    size_t idx = (size_t)blockIdx.x * 256 + threadIdx.x;
    const int D4 = D_ / 4;                 // 192
    int    d4  = (int)(idx % D4);
    size_t row = idx / D4;                 // b*4096 + n
    int    b   = (int)(row >> 12);         // /4096
    v4f g = ((const v4f*)gate)[b * D4 + d4];
    v4f v = x4[idx];
    v *= g;
    __builtin_nontemporal_store(v, &out4[idx]);   // global_store_b128 TH=NT
}

// ---------------------------------------------------------------------------
extern "C" void kernel_launch(void* const* d_in, const int* in_sizes, int n_in,
                              void* d_out, int out_size, void* d_ws, size_t ws_size,
                              hipStream_t stream) {
    const float* x      = (const float*)d_in[0];   // [8,4096,768]
    const float* wht_w  = (const float*)d_in[1];   // [1024]
    const float* gate_w = (const float*)d_in[2];   // [768,1024]
    const float* gate_b = (const float*)d_in[3];   // [768]
    float* out = (float*)d_out;                    // [8,4096,768]

    // workspace layout (floats): c[8*768] | cw[16*1024] | gate[8*768]
    float* c    = (float*)d_ws;
    float* cw   = c + B_ * D_;
    float* gate = cw + 16 * P_;

    sewht_zero_kernel<<<(B_ * D_ + 255) / 256, 256, 0, stream>>>(c);

    dim3 gmean(3, 16, 8);                  // d-chunks, n-chunks, batches
    sewht_mean_kernel<<<gmean, 256, 0, stream>>>(x, c);

    sewht_wht_kernel<<<1, 1024, 0, stream>>>(c, wht_w, cw);

    sewht_gate_kernel<<<D_ / 16, 32, 0, stream>>>(cw, gate_w, gate_b, gate);

    sewht_scale_kernel<<<(B_ * N_ * (D_ / 4)) / 256, 256, 0, stream>>>(
        (const v4f*)x, gate, (v4f*)out);
}